// CollatedVanillaCNN_24292335026552
// MI455X (gfx1250) — compile-verified
//
#include <hip/hip_runtime.h>
#include <math.h>

// CDNA5 / gfx1250: wave32, fp32 WMMA 16x16x4 (lossless fp32 matrix path).
typedef float v2f __attribute__((ext_vector_type(2)));
typedef float v8f __attribute__((ext_vector_type(8)));

#define WMMA_F32(a, b, c) \
  __builtin_amdgcn_wmma_f32_16x16x4_f32(false, (a), false, (b), (short)0, (c), false, false)

__device__ __forceinline__ v2f ld2g(const float* __restrict__ p) {
  v2f r; r.x = p[0]; r.y = p[1]; return r;
}
__device__ __forceinline__ v2f ld2s(const float* p) {
  v2f r; r.x = p[0]; r.y = p[1]; return r;
}

// One wave = 16 instances (16 consecutive x of one row). Block = 64 threads = 2 waves.
// Grid = 2048 blocks -> 4096 waves = 65536 instances.
__global__ __launch_bounds__(64)
void collated_cnn_wmma(const float* __restrict__ x,
                       const float* __restrict__ c1w,  const float* __restrict__ c1bias,
                       const float* __restrict__ bn1g, const float* __restrict__ bn1b,
                       const float* __restrict__ bn1m, const float* __restrict__ bn1v,
                       const float* __restrict__ c2w,  const float* __restrict__ c2bias,
                       const float* __restrict__ bn2g, const float* __restrict__ bn2b,
                       const float* __restrict__ bn2m, const float* __restrict__ bn2v,
                       const float* __restrict__ f1w,  const float* __restrict__ f1b,
                       const float* __restrict__ bn3g, const float* __restrict__ bn3b,
                       const float* __restrict__ bn3m, const float* __restrict__ bn3v,
                       const float* __restrict__ f2w,  const float* __restrict__ f2b,
                       float* __restrict__ out)
{
  __shared__ float s_sc[112];        // fused BN scales: [0,32)=conv1 [32,96)=conv2 [96,112)=fc1
  __shared__ float s_bi[112];        // fused biases (conv bias + BN folded)
  __shared__ float s_a1t[5760];      // conv1 weights, k'=tap*16+ic order: [(tap*32+m)*20 + ic]
  __shared__ float s_buf[2][3632];   // per wave: [0,1520)=win strip (alias c2b[0,1088)+y1b[1088,1408))
                                     //           [1520,3632)=c1b (n-stride 132)

  const int tid = threadIdx.x;

  // ---- fold BN params (block-shared) ----
  if (tid < 32) { float s = bn1g[tid] * rsqrtf(bn1v[tid] + 1e-5f);
                  s_sc[tid]      = s; s_bi[tid]      = (c1bias[tid] - bn1m[tid]) * s + bn1b[tid]; }
  { float s = bn2g[tid] * rsqrtf(bn2v[tid] + 1e-5f);   // tid < 64 always
    s_sc[32 + tid] = s; s_bi[32 + tid] = (c2bias[tid] - bn2m[tid]) * s + bn2b[tid]; }
  if (tid < 16) { float s = bn3g[tid] * rsqrtf(bn3v[tid] + 1e-5f);
                  s_sc[96 + tid] = s; s_bi[96 + tid] = (f1b[tid]   - bn3m[tid]) * s + bn3b[tid]; }

  // ---- transpose conv1 weights into LDS: a1t[(tap*32+m)*20 + ic] = w[m][ic][tap] ----
  for (int t = tid; t < 4608; t += 64) {
    int ic = t & 15, rm = t >> 4;          // rm = tap*32 + m
    int m = rm & 31, tap = rm >> 5;
    s_a1t[rm * 20 + ic] = c1w[m * 144 + ic * 9 + tap];
  }

  const int w    = tid >> 5;
  const int l    = tid & 31;
  const int n    = l & 15;                 // matrix column / instance
  const int koff = (l >> 4) << 1;          // lanes >=16 hold K+2
  const int ro   = (l >> 4) << 3;          // lanes >=16 hold M+8 rows of C/D

  const int gid = blockIdx.x * 2 + w;      // 0..4095
  const int b   = gid >> 10;
  const int rem = gid & 1023;
  const int y   = rem >> 3;
  const int x0  = (rem & 7) << 4;

  float* wb  = &s_buf[w][0];               // [wi(4)][col(19)][ic pitch 20]
  float* c1b = wb + 1520;                  // [n pitch 132][k' = ic*4+wi*2+wj (128)]
  float* c2b = wb;                         // [n pitch 68][64]   (aliases dead window)
  float* y1b = wb + 1088;                  // [n pitch 20][16]

  __syncthreads();

  // ---- stage log1p window strip: 16ch x 4 rows x 19 cols; ic innermost (pitch 20) ----
  for (int t = l; t < 1216; t += 32) {     // 1216 = 32*38, no divergence
    int ic  = t / 76;
    int r2  = t - ic * 76;
    int wi  = r2 / 19;
    int col = r2 - wi * 19;
    int gy = y + wi, gx = x0 + col;
    float v = 0.0f;
    if (gy < 128 && gx < 128)              // right/bottom zero padding
      v = log1pf(x[((b * 16 + ic) * 128 + gy) * 128 + gx]);
    wb[(wi * 19 + col) * 20 + ic] = v;
  }
  __syncthreads();

  // ---- Conv1: K'=tap*16+ic, only valid taps (scalar bounds, no masking) ----
#pragma unroll 1
  for (int g = 0; g < 4; ++g) {            // pooled position
    const int gi = g >> 1, gj = g & 1;
    v8f pl0 = {}, pl1 = {};
#pragma unroll 1
    for (int s = 0; s < 4; ++s) {          // sub-position in pool window
      const int pi = gi * 2 + (s >> 1);
      const int pj = gj * 2 + (s & 1);
      v8f acc0 = {}, acc1 = {};
      const int dloi = (pi == 0) ? 1 : 0, dhii = (pi == 3) ? 1 : 2;
      const int dloj = (pj == 0) ? 1 : 0, dhij = (pj == 3) ? 1 : 2;
#pragma unroll 1
      for (int di = dloi; di <= dhii; ++di) {
        const int wi = pi + di - 1;
#pragma unroll 1
        for (int dj = dloj; dj <= dhij; ++dj) {
          const int wj  = pj + dj - 1;
          const int tap = di * 3 + dj;
          const float* bp  = wb    + (wi * 19 + n + wj) * 20 + koff;
          const float* ap0 = s_a1t + (tap * 32 + n) * 20 + koff;
          const float* ap1 = ap0 + 16 * 20;
#pragma unroll
          for (int qq = 0; qq < 4; ++qq) { // 4 ic-chunks of K=4
            v2f bf = ld2s(bp  + qq * 4);
            v2f a0 = ld2s(ap0 + qq * 4);
            v2f a1 = ld2s(ap1 + qq * 4);
            acc0 = WMMA_F32(a0, bf, acc0);
            acc1 = WMMA_F32(a1, bf, acc1);
          }
        }
      }
#pragma unroll
      for (int r = 0; r < 8; ++r) {        // fused BN + ReLU, accumulate pool
        int m0 = ro + r, m1 = 16 + ro + r;
        pl0[r] += fmaxf(0.f, acc0[r] * s_sc[m0] + s_bi[m0]);
        pl1[r] += fmaxf(0.f, acc1[r] * s_sc[m1] + s_bi[m1]);
      }
    }
#pragma unroll
    for (int r = 0; r < 8; ++r) {          // mean pool -> c1b (pitch 132: conflict-free)
      c1b[n * 132 + (ro + r) * 4 + g]      = pl0[r] * 0.25f;
      c1b[n * 132 + (16 + ro + r) * 4 + g] = pl1[r] * 0.25f;
    }
  }
  __syncthreads();

  // ---- Conv2: K'=ic*4+wi*2+wj (only the 4 live taps of the padded 2x2 input) ----
  {
    v8f p2[4] = {};
    const int wi2 = l >> 4;                // lane-constant wi
#pragma unroll 1
    for (int p = 0; p < 4; ++p) {
      const int pi = p >> 1, pj = p & 1;
      const int wbase = (wi2 + 1 - pi) * 3 + (1 - pj);   // weight (di,dj) base, +j contiguous
      const float* a0p = c2w + n * 288 + wbase;
      const float* a1p = a0p + 16 * 288;
      const float* a2p = a0p + 32 * 288;
      const float* a3p = a0p + 48 * 288;
      const float* bp  = c1b + n * 132 + koff;
      v8f cc0 = {}, cc1 = {}, cc2 = {}, cc3 = {};
#pragma unroll 4
      for (int q = 0; q < 32; ++q) {       // ic chunks
        v2f bf = ld2s(bp + q * 4);
        v2f a0 = ld2g(a0p + q * 9);
        v2f a1 = ld2g(a1p + q * 9);
        v2f a2 = ld2g(a2p + q * 9);
        v2f a3 = ld2g(a3p + q * 9);
        cc0 = WMMA_F32(a0, bf, cc0);
        cc1 = WMMA_F32(a1, bf, cc1);
        cc2 = WMMA_F32(a2, bf, cc2);
        cc3 = WMMA_F32(a3, bf, cc3);
      }
#pragma unroll
      for (int r = 0; r < 8; ++r) {
        int mB = ro + r;
        p2[0][r] += fmaxf(0.f, cc0[r] * s_sc[32 + mB] + s_bi[32 + mB]);
        p2[1][r] += fmaxf(0.f, cc1[r] * s_sc[48 + mB] + s_bi[48 + mB]);
        p2[2][r] += fmaxf(0.f, cc2[r] * s_sc[64 + mB] + s_bi[64 + mB]);
        p2[3][r] += fmaxf(0.f, cc3[r] * s_sc[80 + mB] + s_bi[80 + mB]);
      }
    }
#pragma unroll
    for (int h = 0; h < 4; ++h)
#pragma unroll
      for (int r = 0; r < 8; ++r)
        c2b[n * 68 + h * 16 + ro + r] = p2[h][r] * 0.25f;  // mean pool 2x2 -> 1x1
  }
  __syncthreads();

  // ---- FC1: M=16, K=64, fused BN3 + ReLU ----
  {
    v8f cf = {};
    const float* bp = c2b + n * 68 + koff;
    const float* ap = f1w + n * 64 + koff;
#pragma unroll
    for (int q = 0; q < 16; ++q) {
      v2f bf = ld2s(bp + q * 4);
      v2f a  = ld2g(ap + q * 4);
      cf = WMMA_F32(a, bf, cf);
    }
#pragma unroll
    for (int r = 0; r < 8; ++r) {
      int m = ro + r;
      y1b[n * 20 + m] = fmaxf(0.f, cf[r] * s_sc[96 + m] + s_bi[96 + m]);
    }
  }
  __syncthreads();

  // ---- FC2: M=8 (A rows 8..15 zero), K=16; expm1; store ----
  {
    v8f co = {};
    const float* bp = y1b + n * 20 + koff;
#pragma unroll
    for (int q = 0; q < 4; ++q) {
      v2f bf = ld2s(bp + q * 4);
      v2f a;
      if (n < 8) a = ld2g(f2w + n * 16 + q * 4 + koff);
      else { a.x = 0.f; a.y = 0.f; }
      co = WMMA_F32(a, bf, co);
    }
    if (l < 16) {                          // valid D rows M=0..7 live in lanes 0..15
#pragma unroll
      for (int r = 0; r < 8; ++r) {
        float v = expm1f(co[r] + f2b[r]);
        out[((b * 8 + r) * 128 + y) * 128 + x0 + n] = v;
      }
    }
  }
}

extern "C" void kernel_launch(void* const* d_in, const int* in_sizes, int n_in,
                              void* d_out, int out_size, void* d_ws, size_t ws_size,
                              hipStream_t stream) {
  (void)in_sizes; (void)n_in; (void)out_size; (void)d_ws; (void)ws_size;
  const float* x    = (const float*)d_in[0];
  const float* c1w  = (const float*)d_in[1];
  const float* c1bb = (const float*)d_in[2];
  const float* bn1g = (const float*)d_in[3];
  const float* bn1b = (const float*)d_in[4];
  const float* bn1m = (const float*)d_in[5];
  const float* bn1v = (const float*)d_in[6];
  const float* c2w  = (const float*)d_in[7];
  const float* c2bb = (const float*)d_in[8];
  const float* bn2g = (const float*)d_in[9];
  const float* bn2b = (const float*)d_in[10];
  const float* bn2m = (const float*)d_in[11];
  const float* bn2v = (const float*)d_in[12];
  const float* f1w  = (const float*)d_in[13];
  const float* f1b  = (const float*)d_in[14];
  const float* bn3g = (const float*)d_in[15];
  const float* bn3b = (const float*)d_in[16];
  const float* bn3m = (const float*)d_in[17];
  const float* bn3v = (const float*)d_in[18];
  const float* f2w  = (const float*)d_in[19];
  const float* f2b  = (const float*)d_in[20];

  collated_cnn_wmma<<<2048, 64, 0, stream>>>(
      x, c1w, c1bb, bn1g, bn1b, bn1m, bn1v,
      c2w, c2bb, bn2g, bn2b, bn2m, bn2v,
      f1w, f1b, bn3g, bn3b, bn3m, bn3v,
      f2w, f2b, (float*)d_out);
}